// OfflineSlidingWindowAttn_60902636258019
// MI455X (gfx1250) — compile-verified
//
#include <hip/hip_runtime.h>
#include <hip/hip_bf16.h>
#include <math.h>

typedef __attribute__((ext_vector_type(2))) float v2f;
typedef __attribute__((ext_vector_type(8))) float v8f;

#define HEAD_DIM   128
#define NUM_Q_HEAD 16
#define NUM_KV_HEAD 4
#define WINDOW     256
#define SCALE      0.08838834764831845f   // 1/sqrt(128)
#define CAP        30.0f
#define INV_CAP    (1.0f/30.0f)
#define CLIP_MIN   (-0.03f)
#define CLIP_LEN   1.06f

#define QK_STRIDE  132   // LDS row stride for 128-wide tiles (conflict-free)
#define S_STRIDE   292   // LDS row stride for logits buffer (<=272 cols used)

// One wave (32 threads) per (16-query tile, q-head).
__launch_bounds__(32)
__global__ void OfflineSlidingWindowAttn_60902636258019_kernel(
    const float* __restrict__ q, const float* __restrict__ k,
    const float* __restrict__ v, float* __restrict__ out, int S)
{
    const int qt   = blockIdx.x;      // query tile index
    const int h    = blockIdx.y;      // q head
    const int kh   = h >> 2;          // kv head (repeat_interleave by 4)
    const int i0   = qt * 16;
    const int lane = threadIdx.x;     // 0..31 (wave32)
    const int m    = lane & 15;
    const int hi   = lane >> 4;       // 0 = lanes 0-15, 1 = lanes 16-31

    __shared__ __align__(16) float ldsQ[16 * QK_STRIDE];
    __shared__ __align__(16) float ldsK[16 * QK_STRIDE];
    __shared__ __align__(16) float ldsV[16 * QK_STRIDE];
    __shared__ __align__(16) float ldsS[16 * S_STRIDE];
    __shared__ float rowM[16];
    __shared__ float rowInvD[16];

    // Key-tile range covering [max(0, i0-WINDOW), i0+15]; tiles are 16-aligned.
    int jlo = i0 - WINDOW; if (jlo < 0) jlo = 0;
    const int t0    = jlo >> 4;
    const int t1    = i0  >> 4;       // inclusive
    const int ncols = (t1 - t0 + 1) * 16;

    // ---- Stage Q tile (16 x 128) into LDS ----
    for (int idx = lane; idx < 16 * 32; idx += 32) {
        const int r = idx >> 5, c4 = (idx & 31) << 2;
        *(float4*)&ldsQ[r * QK_STRIDE + c4] =
            *(const float4*)&q[((size_t)(i0 + r) * NUM_Q_HEAD + h) * HEAD_DIM + c4];
    }
    __syncthreads();

    // ================= Pass 1: S = scale * Q K^T, tanh cap, mask =================
    for (int t = t0; t <= t1; ++t) {
        const int j0 = t << 4;
        for (int idx = lane; idx < 16 * 32; idx += 32) {
            const int r = idx >> 5, c4 = (idx & 31) << 2;
            *(float4*)&ldsK[r * QK_STRIDE + c4] =
                *(const float4*)&k[((size_t)(j0 + r) * NUM_KV_HEAD + kh) * HEAD_DIM + c4];
        }
        __syncthreads();

        v8f acc = {0.f, 0.f, 0.f, 0.f, 0.f, 0.f, 0.f, 0.f};
        const int off = hi * 2;  // upper half-wave holds K= +2,+3 of each chunk
        #pragma unroll
        for (int c = 0; c < 32; ++c) {
            const int kb = c * 4 + off;
            v2f a = *(const v2f*)&ldsQ[m * QK_STRIDE + kb];  // A: Q[16 x 4]
            v2f b = *(const v2f*)&ldsK[m * QK_STRIDE + kb];  // B: K^T[4 x 16]
            acc = __builtin_amdgcn_wmma_f32_16x16x4_f32(
                false, a, false, b, (short)0, acc, false, false);
        }

        // Post-process D (lane=N, vgpr r = row M + 8*hi) -> LDS logits
        const int colbase = (t - t0) * 16;
        #pragma unroll
        for (int r = 0; r < 8; ++r) {
            const int M = r + hi * 8;
            const int i = i0 + M;
            const int j = j0 + m;
            float x = acc[r] * SCALE;
            x = CAP * tanhf(x * INV_CAP);
            const bool allowed = (j <= i) && (i - j <= WINDOW);
            ldsS[M * S_STRIDE + colbase + m] = allowed ? x : -__builtin_inff();
        }
        __syncthreads();
    }

    // ================= Row max / denom (diagonal always allowed => denom > 0) ====
    if (lane < 16) {
        const int r = lane;
        float mx = -__builtin_inff();
        for (int c = 0; c < ncols; ++c) mx = fmaxf(mx, ldsS[r * S_STRIDE + c]);
        float s = 0.f;
        for (int c = 0; c < ncols; ++c) s += __expf(ldsS[r * S_STRIDE + c] - mx);
        rowM[r]    = mx;
        rowInvD[r] = 1.0f / s;
    }
    __syncthreads();

    // ================= Form clipped probabilities in place ======================
    for (int r = 0; r < 16; ++r) {
        const float mx  = rowM[r];
        const float inv = rowInvD[r];
        for (int c = lane; c < ncols; c += 32) {
            const float e = __expf(ldsS[r * S_STRIDE + c] - mx);  // exp(-inf)=0 for masked
            float p = CLIP_LEN * e * inv + CLIP_MIN;
            p = fminf(fmaxf(p, 0.0f), 1.0f);
            ldsS[r * S_STRIDE + c] = p;
        }
    }
    __syncthreads();

    // ================= Pass 2: O = P V ==========================================
    v8f o[8];
    #pragma unroll
    for (int n = 0; n < 8; ++n) o[n] = (v8f){0.f, 0.f, 0.f, 0.f, 0.f, 0.f, 0.f, 0.f};

    for (int t = t0; t <= t1; ++t) {
        const int j0 = t << 4;
        for (int idx = lane; idx < 16 * 32; idx += 32) {
            const int r = idx >> 5, c4 = (idx & 31) << 2;
            *(float4*)&ldsV[r * QK_STRIDE + c4] =
                *(const float4*)&v[((size_t)(j0 + r) * NUM_KV_HEAD + kh) * HEAD_DIM + c4];
        }
        __syncthreads();

        const int colbase = (t - t0) * 16;
        #pragma unroll
        for (int kc = 0; kc < 4; ++kc) {
            const int kb = kc * 4 + hi * 2;  // local key row within this 16-key tile
            v2f a = *(const v2f*)&ldsS[m * S_STRIDE + colbase + kb];  // A: P[16 x 4]
            #pragma unroll
            for (int n = 0; n < 8; ++n) {
                const int n0 = n * 16;
                v2f b;                                        // B: V[4 x 16]
                b.x = ldsV[(kb + 0) * QK_STRIDE + n0 + m];
                b.y = ldsV[(kb + 1) * QK_STRIDE + n0 + m];
                o[n] = __builtin_amdgcn_wmma_f32_16x16x4_f32(
                    false, a, false, b, (short)0, o[n], false, false);
            }
        }
        __syncthreads();
    }

    // ---- Store O tile (16 x 128) ----
    #pragma unroll
    for (int n = 0; n < 8; ++n) {
        #pragma unroll
        for (int r = 0; r < 8; ++r) {
            const int M = r + hi * 8;
            out[((size_t)(i0 + M) * NUM_Q_HEAD + h) * HEAD_DIM + n * 16 + m] = o[n][r];
        }
    }
}

extern "C" void kernel_launch(void* const* d_in, const int* in_sizes, int n_in,
                              void* d_out, int out_size, void* d_ws, size_t ws_size,
                              hipStream_t stream) {
    const float* q = (const float*)d_in[0];
    const float* k = (const float*)d_in[1];
    const float* v = (const float*)d_in[2];
    float* out = (float*)d_out;
    const int S = in_sizes[0] / (NUM_Q_HEAD * HEAD_DIM);  // 2048
    dim3 grid(S / 16, NUM_Q_HEAD);
    OfflineSlidingWindowAttn_60902636258019_kernel<<<grid, 32, 0, stream>>>(q, k, v, out, S);
}